// DydenseNet_89653147337269
// MI455X (gfx1250) — compile-verified
//
#include <hip/hip_runtime.h>

typedef float v2f __attribute__((ext_vector_type(2)));
typedef float v8f __attribute__((ext_vector_type(8)));

#define BN_SCALE 0.9999950000374997f   // 1/sqrt(1 + 1e-5)
#define NHEADS 4

// ---------------- stem: 3x3x3 conv, stride 2, 1->16 channels ----------------
__global__ void stem_conv(const float* __restrict__ x, const float* __restrict__ w,
                          float* __restrict__ out) {
  // in: (16,1,200,15,15); out: (16,16,100,8,8) into buf0 (channel capacity 128)
  const int S = 100 * 8 * 8;
  int id = blockIdx.x * blockDim.x + threadIdx.x;
  if (id >= 16 * 16 * S) return;
  int s  = id % S;
  int oc = (id / S) % 16;
  int b  = id / (S * 16);
  int d = s / 64, h = (s / 8) % 8, ww = s % 8;
  const float* xb = x + (size_t)b * 200 * 15 * 15;
  float acc = 0.f;
  for (int kd = 0; kd < 3; ++kd) {
    int di = 2 * d + kd - 1;
    if (di < 0 || di >= 200) continue;
    for (int kh = 0; kh < 3; ++kh) {
      int hi = 2 * h + kh - 1;
      if (hi < 0 || hi >= 15) continue;
      for (int kw = 0; kw < 3; ++kw) {
        int wi = 2 * ww + kw - 1;
        if (wi < 0 || wi >= 15) continue;
        acc += xb[(di * 15 + hi) * 15 + wi] * w[oc * 27 + kd * 9 + kh * 3 + kw];
      }
    }
  }
  out[((size_t)b * 128 + oc) * S + s] = acc;
}

// ---------------- per-(b,c) spatial mean of relu(bn1(x)) ----------------
__global__ void avg_bnrelu(const float* __restrict__ xin, const float* __restrict__ g,
                           const float* __restrict__ bb, float* __restrict__ avg,
                           int C, int Cmax, int S) {
  int c = blockIdx.x, b = blockIdx.y;
  const float* xp = xin + ((size_t)b * Cmax + c) * S;
  float sc = g[c] * BN_SCALE, bi = bb[c];
  float sum = 0.f;
  for (int s = threadIdx.x; s < S; s += blockDim.x)
    sum += fmaxf(xp[s] * sc + bi, 0.f);
  __shared__ float red[256];
  red[threadIdx.x] = sum;
  __syncthreads();
  for (int off = 128; off > 0; off >>= 1) {
    if (threadIdx.x < off) red[threadIdx.x] += red[threadIdx.x + off];
    __syncthreads();
  }
  if (threadIdx.x == 0) avg[b * C + c] = red[0] / (float)S;
}

// ---------------- SE gate + exact k-th-smallest pruning ----------------
__global__ void gate(const float* __restrict__ avg, const float* __restrict__ fc1,
                     const float* __restrict__ fc2w, const float* __restrict__ fc2b,
                     float* __restrict__ mask, int C, int kdim, int kth) {
  int h = blockIdx.x, b = blockIdx.y;
  int t = threadIdx.x;
  __shared__ float sy[64];
  __shared__ float sm[800];
  __shared__ float sthr;
  if (t < kdim) {
    const float* f1 = fc1 + (size_t)(h * kdim + t) * C;
    const float* av = avg + b * C;
    float acc = 0.f;
    for (int c = 0; c < C; ++c) acc += f1[c] * av[c];
    sy[t] = fmaxf(acc, 0.f);
  }
  __syncthreads();
  for (int c = t; c < C; c += blockDim.x) {
    const float* f2 = fc2w + (size_t)(h * C + c) * kdim;
    float acc = fc2b[h * C + c];
    for (int k = 0; k < kdim; ++k) acc += f2[k] * sy[k];
    sm[c] = fmaxf(acc, 0.f);
  }
  __syncthreads();
  // thr = kth smallest (rank counting handles ties; matches sort()[k-1])
  for (int c = t; c < C; c += blockDim.x) {
    float v = sm[c];
    int lt = 0, le = 0;
    for (int j = 0; j < C; ++j) { lt += (sm[j] < v); le += (sm[j] <= v); }
    if (lt < kth && le >= kth) sthr = v;
  }
  __syncthreads();
  float thr = sthr;
  for (int c = t; c < C; c += blockDim.x) {
    float v = sm[c];
    mask[(size_t)(b * NHEADS + h) * C + c] = (v <= thr) ? 0.f : v;
  }
}

// -------- fused dynamic 1x1x1 conv: bn1+relu -> (mask*W) GEMM -> shuffle+bn2+relu --------
// out1 x C GEMM vs C x S, per batch, using V_WMMA_F32_16X16X4_F32.
// Branchless K-loop: out-of-range columns are clamped (they compute garbage in
// their own output column only, which is never stored) so EXEC stays all-ones.
__global__ void __launch_bounds__(128)
gemm1x1_wmma(const float* __restrict__ xin, const float* __restrict__ w1,
             const float* __restrict__ mask, const float* __restrict__ bn1g,
             const float* __restrict__ bn1b, const float* __restrict__ bn2g,
             const float* __restrict__ bn2b, float* __restrict__ h2,
             int C, int Cmax, int S, int out1) {
  __shared__ float As[16 * 769];   // A' tile, padded stride C+1
  __shared__ float sS1[800];
  __shared__ float sB1[800];
  const int b  = blockIdx.z;
  const int m0 = blockIdx.y * 16;
  const int o  = out1 >> 2;        // per-head out channels
  const int tid = threadIdx.x;
  const int Cp = C + 1;

  for (int c = tid; c < C; c += 128) {
    sS1[c] = bn1g[c] * BN_SCALE;
    sB1[c] = bn1b[c];
  }
  // A'[mr,c] = mask[b, h(m), c] * conv1_w[m, c]   (m = h*o + oi, dense over c)
  for (int idx = tid; idx < 16 * C; idx += 128) {
    int mr = idx / C, c = idx - mr * C;
    int m = m0 + mr;
    int hh = m / o;
    As[mr * Cp + c] = w1[(size_t)m * C + c] * mask[(size_t)(b * NHEADS + hh) * C + c];
  }
  __syncthreads();

  const int wave = tid >> 5;          // 4 waves, each a 16-wide N sub-tile
  const int lane = tid & 31;
  const int half = lane >> 4;         // 0: K={k0,k0+1}, 1: K={k0+2,k0+3}
  const int am   = lane & 15;         // A-fragment row (M)
  const int n0   = blockIdx.x * 64 + wave * 16 + (lane & 15);
  const bool nvalid = (n0 < S);
  const int nc = nvalid ? n0 : (S - 1);            // clamp: keep loads in-bounds

  const float* xk = xin + (size_t)b * Cmax * S + (size_t)(2 * half) * S + nc;
  const float* ap = As + am * Cp + 2 * half;
  const size_t xstep = (size_t)4 * S;

  v8f acc = {};
  for (int k0 = 0; k0 < C; k0 += 4) {
    v2f a, bf;
    a[0] = ap[0];
    a[1] = ap[1];
    int ka = k0 + 2 * half;
    float v0 = xk[0];
    float v1 = xk[S];
    bf[0] = fmaxf(v0 * sS1[ka] + sB1[ka], 0.f);
    bf[1] = fmaxf(v1 * sS1[ka + 1] + sB1[ka + 1], 0.f);
    // (neg_a, A, neg_b, B, c_mod, C, reuse_a, reuse_b)
    acc = __builtin_amdgcn_wmma_f32_16x16x4_f32(false, a, false, bf,
                                                (short)0, acc, false, false);
    ap += 4;
    xk += xstep;
  }
  if (nvalid) {
#pragma unroll
    for (int v = 0; v < 8; ++v) {
      int m = m0 + v + 8 * half;               // C/D layout: M = v + 8*(lane/16)
      int oi = m % o, hh = m / o;
      int c2 = oi * NHEADS + hh;               // channel shuffle
      float val = acc[v] * (bn2g[c2] * BN_SCALE) + bn2b[c2];
      h2[((size_t)b * out1 + c2) * S + n0] = fmaxf(val, 0.f);
    }
  }
}

// ---------------- grouped 3x3x3 conv (groups=4), writes new dense channels ----------------
__global__ void conv2_grouped(const float* __restrict__ h2, const float* __restrict__ w2,
                              float* __restrict__ xout, int g, int D, int H, int W,
                              int Cmax, int Cofs) {
  int S = D * H * W;
  int id = blockIdx.x * blockDim.x + threadIdx.x;
  if (id >= 16 * g * S) return;
  int s  = id % S;
  int oc = (id / S) % g;
  int b  = id / (S * g);
  int d = s / (H * W), hh = (s / W) % H, ww = s % W;
  int grp = oc / (g >> 2);                    // g/4 out channels per group
  const float* hb = h2 + ((size_t)b * 4 * g + grp * g) * S;
  const float* wp = w2 + (size_t)oc * g * 27;
  float acc = 0.f;
  for (int ic = 0; ic < g; ++ic) {
    const float* hc = hb + (size_t)ic * S;
    const float* wc = wp + ic * 27;
    for (int kd = 0; kd < 3; ++kd) {
      int di = d + kd - 1; if (di < 0 || di >= D) continue;
      for (int kh = 0; kh < 3; ++kh) {
        int hi = hh + kh - 1; if (hi < 0 || hi >= H) continue;
        for (int kw = 0; kw < 3; ++kw) {
          int wi = ww + kw - 1; if (wi < 0 || wi >= W) continue;
          acc += hc[(di * H + hi) * W + wi] * wc[kd * 9 + kh * 3 + kw];
        }
      }
    }
  }
  xout[((size_t)b * Cmax + Cofs + oc) * S + s] = acc;
}

// ---------------- 2x2x2 average pool ----------------
__global__ void pool2(const float* __restrict__ xin, float* __restrict__ xout,
                      int C, int CmaxIn, int CmaxOut, int Do, int Ho, int Wo) {
  int So = Do * Ho * Wo;
  int id = blockIdx.x * blockDim.x + threadIdx.x;
  if (id >= 16 * C * So) return;
  int s = id % So; int c = (id / So) % C; int b = id / (So * C);
  int d = s / (Ho * Wo), h = (s / Wo) % Ho, w = s % Wo;
  int Hi = 2 * Ho, Wi = 2 * Wo;
  const float* xp = xin + ((size_t)b * CmaxIn + c) * (size_t)(2 * Do * Hi * Wi);
  float sum = 0.f;
  for (int kd = 0; kd < 2; ++kd)
    for (int kh = 0; kh < 2; ++kh)
      for (int kw = 0; kw < 2; ++kw)
        sum += xp[((2 * d + kd) * Hi + 2 * h + kh) * Wi + 2 * w + kw];
  xout[((size_t)b * CmaxOut + c) * So + s] = sum * 0.125f;
}

// ---------------- bn_last + relu + global mean ----------------
__global__ void final_feat(const float* __restrict__ xin, const float* __restrict__ g,
                           const float* __restrict__ bb, float* __restrict__ feat) {
  int id = blockIdx.x * blockDim.x + threadIdx.x;   // 16*800
  if (id >= 16 * 800) return;
  int c = id % 800, b = id / 800;
  const float* xp = xin + ((size_t)b * 800 + c) * 100;
  float sc = g[c] * BN_SCALE, bi = bb[c];
  float sum = 0.f;
  for (int s = 0; s < 100; ++s) sum += fmaxf(xp[s] * sc + bi, 0.f);
  feat[id] = sum * 0.01f;
}

// ---------------- classifier 800 -> 16 ----------------
__global__ void classifier(const float* __restrict__ feat, const float* __restrict__ w,
                           const float* __restrict__ bias, float* __restrict__ out) {
  int t = threadIdx.x;                 // 256 threads: (b, n)
  int n = t % 16, b = t / 16;
  const float* fb = feat + b * 800;
  const float* wn = w + n * 800;
  float acc = bias[n];
  for (int c = 0; c < 800; ++c) acc += fb[c] * wn[c];
  out[b * 16 + n] = acc;
}

extern "C" void kernel_launch(void* const* d_in, const int* in_sizes, int n_in,
                              void* d_out, int out_size, void* d_ws, size_t ws_size,
                              hipStream_t stream) {
  (void)in_sizes; (void)n_in; (void)out_size; (void)ws_size;
  const float* x = (const float*)d_in[0];
  auto P = [&](int i) { return (const float*)d_in[i]; };
  // JAX tree_flatten order (dict keys sorted): blocks leaves 1..378, then
  // bn_last_b=379, bn_last_g=380, cls_b=381, cls_w=382, init_conv=383.
  // per-layer: +0 bn1_b, +1 bn1_g, +2 bn2_b, +3 bn2_g, +4 conv1_w,
  //            +5 conv2_w, +6 fc1_w, +7 fc2_b, +8 fc2_w

  float* ws = (float*)d_ws;
  float* buf0  = ws;
  float* buf1  = buf0 + (size_t)16 * 128 * 6400;
  float* buf2  = buf1 + (size_t)16 * 352 * 800;
  float* h2    = buf2 + (size_t)16 * 800 * 100;
  float* avgb  = h2 + (size_t)16 * 32 * 6400;
  float* maskb = avgb + 16 * 800;
  float* featb = maskb + 16 * 4 * 800;

  { // stem
    int total = 16 * 16 * 6400;
    stem_conv<<<(total + 255) / 256, 256, 0, stream>>>(x, P(383), buf0);
  }

  const int growth[3] = {8, 16, 32};
  const int Dd[3] = {100, 50, 25}, Hh[3] = {8, 4, 2}, Wd[3] = {8, 4, 2};
  const int Cmax[3] = {128, 352, 800};
  float* bufs[3] = {buf0, buf1, buf2};
  int C = 16;
  for (int i = 0; i < 3; ++i) {
    int S = Dd[i] * Hh[i] * Wd[i];
    int g = growth[i], out1 = 4 * g;
    for (int j = 0; j < 14; ++j) {
      int base = 1 + (i * 14 + j) * 9;
      const float* bn1_b = P(base + 0);
      const float* bn1_g = P(base + 1);
      const float* bn2_b = P(base + 2);
      const float* bn2_g = P(base + 3);
      const float* conv1 = P(base + 4);
      const float* conv2w = P(base + 5);
      const float* fc1  = P(base + 6);
      const float* fc2b = P(base + 7);
      const float* fc2w = P(base + 8);
      int sqe  = (C < 80) ? 8 : 16;
      int kdim = C / sqe;
      int kth  = (3 * C) / 4;        // round(C*0.75), C % 8 == 0

      dim3 ga(C, 16);
      avg_bnrelu<<<ga, 256, 0, stream>>>(bufs[i], bn1_g, bn1_b, avgb, C, Cmax[i], S);
      dim3 gg(NHEADS, 16);
      gate<<<gg, 256, 0, stream>>>(avgb, fc1, fc2w, fc2b, maskb, C, kdim, kth);
      dim3 gm((S + 63) / 64, out1 / 16, 16);
      gemm1x1_wmma<<<gm, 128, 0, stream>>>(bufs[i], conv1, maskb, bn1_g, bn1_b,
                                           bn2_g, bn2_b, h2, C, Cmax[i], S, out1);
      int total = 16 * g * S;
      conv2_grouped<<<(total + 255) / 256, 256, 0, stream>>>(
          h2, conv2w, bufs[i], g, Dd[i], Hh[i], Wd[i], Cmax[i], C);
      C += g;
    }
    if (i < 2) {
      int So = Dd[i + 1] * Hh[i + 1] * Wd[i + 1];
      int total = 16 * C * So;
      pool2<<<(total + 255) / 256, 256, 0, stream>>>(
          bufs[i], bufs[i + 1], C, Cmax[i], Cmax[i + 1], Dd[i + 1], Hh[i + 1], Wd[i + 1]);
    }
  }
  final_feat<<<(16 * 800 + 255) / 256, 256, 0, stream>>>(buf2, P(380), P(379), featb);
  classifier<<<1, 256, 0, stream>>>(featb, P(382), P(381), (float*)d_out);
}